// Model_4922032521878
// MI455X (gfx1250) — compile-verified
//
#include <hip/hip_runtime.h>

typedef __attribute__((ext_vector_type(2))) float v2f;
typedef __attribute__((ext_vector_type(8))) float v8f;

#define NK        10000
#define KSZ       9
#define CIN       3
#define LEN       256
#define GROUPN    2000
#define CH_STRIDE 384          // 64 halo + 256 + 64 halo
#define ZOFF      1152         // 3*384 ; start of guaranteed-zero slab
#define LDSX      (ZOFF + 256) // zero slab covers +tile_base(240)+lane(15)

// ---------------------------------------------------------------------------
// Kernel 1: dilated conv via V_WMMA_F32_16X16X4_F32 + fused max / PPV features
// one wave32 per (batch, 16-kernel tile); 7 K-steps (27 taps padded to 28)
// ---------------------------------------------------------------------------
__global__ __launch_bounds__(32)
void rocket_conv_feat(const float* __restrict__ x_enc,   // (B, L, C)
                      const float* __restrict__ W,       // (NK, C, KS)
                      const float* __restrict__ bias,    // (NK)
                      float* __restrict__ feat)          // (B, 2*NK)
{
    __shared__ float xs[LDSX];
    const int lane  = threadIdx.x;        // 0..31
    const int tile  = blockIdx.x;         // 0..624
    const int batch = blockIdx.y;         // 0..15
    const int kbase = tile * 16;
    const int grp   = kbase / GROUPN;     // dilation group 0..4
    const int dil   = 1 << grp;

    // Stage x[batch] into LDS with zero halo (implements SAME dilated padding)
    for (int idx = lane; idx < LDSX; idx += 32) {
        float v = 0.0f;
        if (idx < ZOFF) {
            int c = idx / CH_STRIDE;
            int p = idx - c * CH_STRIDE;
            if (p >= 64 && p < 64 + LEN)
                v = x_enc[batch * (LEN * CIN) + (p - 64) * CIN + c];
        }
        xs[idx] = v;
    }
    __syncthreads();

    const int hi = (lane >= 16) ? 1 : 0;
    const int ln = lane & 15;

    // A operand (weights): f32 16x4 layout -> lane M=ln, VGPR r: K = 4*kb+r+2*hi
    v2f a[7];
    int boff[7][2];                        // matching LDS base offsets for B rows
    const int mker = kbase + ln;
    #pragma unroll
    for (int kb = 0; kb < 7; ++kb) {
        #pragma unroll
        for (int r = 0; r < 2; ++r) {
            int   kk = 4 * kb + r + 2 * hi;
            float w  = 0.0f;
            int   bo = ZOFF + ln;          // padded K row -> zero slab
            if (kk < 27) {
                w     = W[mker * 27 + kk];
                int c = kk / KSZ;
                int j = kk - c * KSZ;
                bo    = c * CH_STRIDE + 64 + (j - 4) * dil + ln;
            }
            a[kb][r]    = w;
            boff[kb][r] = bo;
        }
    }

    // bias per C/D slot: VGPR r holds row M = r + 8*hi
    float bslot[8];
    #pragma unroll
    for (int r = 0; r < 8; ++r) bslot[r] = bias[kbase + r + 8 * hi];

    float rmax[8], cnt[8];
    #pragma unroll
    for (int r = 0; r < 8; ++r) { rmax[r] = -3.4e38f; cnt[r] = 0.0f; }

    // 16 position tiles of 16 -> L = 256
    for (int t = 0; t < 16; ++t) {
        const int tb = t * 16;
        v8f c = {};
        #pragma unroll
        for (int kb = 0; kb < 7; ++kb) {
            v2f bv;
            bv[0] = xs[boff[kb][0] + tb];
            bv[1] = xs[boff[kb][1] + tb];
            c = __builtin_amdgcn_wmma_f32_16x16x4_f32(
                    false, a[kb], false, bv, (short)0, c, false, false);
        }
        #pragma unroll
        for (int r = 0; r < 8; ++r) {
            float v = c[r] + bslot[r];
            rmax[r] = fmaxf(rmax[r], v);
            cnt[r] += (v > 0.0f) ? 1.0f : 0.0f;
        }
    }

    // reduce over the 16 positions held across each 16-lane half
    #pragma unroll
    for (int r = 0; r < 8; ++r) {
        #pragma unroll
        for (int m = 1; m <= 8; m <<= 1) {
            rmax[r] = fmaxf(rmax[r], __shfl_xor(rmax[r], m, 32));
            cnt[r] += __shfl_xor(cnt[r], m, 32);
        }
    }

    if (ln == 0) {                          // lanes 0 and 16 write 8 kernels each
        #pragma unroll
        for (int r = 0; r < 8; ++r) {
            int k      = kbase + r + 8 * hi;
            int within = k - grp * GROUPN;
            int fmaxi  = grp * (2 * GROUPN) + within;          // max block
            feat[batch * (2 * NK) + fmaxi]          = rmax[r];
            feat[batch * (2 * NK) + fmaxi + GROUPN] = cnt[r] * (1.0f / 256.0f);
        }
    }
}

// ---------------------------------------------------------------------------
// Kernel 2a: batchnorm stats over B=16 -> per-feature affine (A, Bp)
// ---------------------------------------------------------------------------
__global__ __launch_bounds__(256)
void bn_stats(const float* __restrict__ feat,
              const float* __restrict__ gamma,
              const float* __restrict__ beta,
              float* __restrict__ A, float* __restrict__ Bp)
{
    int f = blockIdx.x * blockDim.x + threadIdx.x;
    if (f >= 2 * NK) return;
    float s = 0.0f, s2 = 0.0f;
    #pragma unroll
    for (int b = 0; b < 16; ++b) {
        float v = feat[b * 2 * NK + f];
        s += v; s2 += v * v;
    }
    float mu  = s * (1.0f / 16.0f);
    float var = s2 * (1.0f / 16.0f) - mu * mu;
    float sc  = rsqrtf(var + 1e-5f) * gamma[f];
    A[f]  = sc;
    Bp[f] = beta[f] - mu * sc;
}

// ---------------------------------------------------------------------------
// Kernel 2b: fused normalize + linear head (20000 -> 10), one block per batch
// ---------------------------------------------------------------------------
__global__ __launch_bounds__(256)
void bn_linear(const float* __restrict__ feat,
               const float* __restrict__ A,
               const float* __restrict__ Bp,
               const float* __restrict__ linW,
               const float* __restrict__ linB,
               float* __restrict__ out)
{
    __shared__ float red[256][10];
    const int b   = blockIdx.x;
    const int tid = threadIdx.x;
    float acc[10];
    #pragma unroll
    for (int n = 0; n < 10; ++n) acc[n] = 0.0f;
    for (int f = tid; f < 2 * NK; f += 256) {
        float v = feat[b * 2 * NK + f] * A[f] + Bp[f];
        #pragma unroll
        for (int n = 0; n < 10; ++n) acc[n] += v * linW[f * 10 + n];
    }
    #pragma unroll
    for (int n = 0; n < 10; ++n) red[tid][n] = acc[n];
    __syncthreads();
    for (int s = 128; s > 0; s >>= 1) {
        if (tid < s) {
            #pragma unroll
            for (int n = 0; n < 10; ++n) red[tid][n] += red[tid + s][n];
        }
        __syncthreads();
    }
    if (tid < 10) out[b * 10 + tid] = red[0][tid] + linB[tid];
}

// ---------------------------------------------------------------------------
extern "C" void kernel_launch(void* const* d_in, const int* in_sizes, int n_in,
                              void* d_out, int out_size, void* d_ws, size_t ws_size,
                              hipStream_t stream) {
    const float* x_enc = (const float*)d_in[0];   // (16,256,3)
    const float* W     = (const float*)d_in[1];   // (10000,3,9)
    const float* bias  = (const float*)d_in[2];   // (10000)
    const float* gamma = (const float*)d_in[3];   // (20000)
    const float* beta  = (const float*)d_in[4];   // (20000)
    const float* linW  = (const float*)d_in[5];   // (20000,10)
    const float* linB  = (const float*)d_in[6];   // (10)
    float* out = (float*)d_out;                   // (16,10)

    float* ws   = (float*)d_ws;
    float* feat = ws;                   // 16*20000 = 320000 floats
    float* Aaf  = ws + 320000;          // 20000
    float* Baf  = ws + 340000;          // 20000

    rocket_conv_feat<<<dim3(625, 16), 32, 0, stream>>>(x_enc, W, bias, feat);
    bn_stats<<<(2 * NK + 255) / 256, 256, 0, stream>>>(feat, gamma, beta, Aaf, Baf);
    bn_linear<<<16, 256, 0, stream>>>(feat, Aaf, Baf, linW, linB, out);
}